// GNNEncoder_32478542692807
// MI455X (gfx1250) — compile-verified
//
#include <hip/hip_runtime.h>
#include <hip/hip_bf16.h>
#include <math.h>

// ---------------------------------------------------------------------------
// GATv2 x2 GNN encoder for gfx1250 (MI455X).
// Node-level GEMMs use V_WMMA_F32_16X16X4_F32 (fp32 matrix pipe, K=4 native).
// Edge-level softmax-aggregation uses float atomics (L2 atomic units).
// ---------------------------------------------------------------------------

typedef __attribute__((ext_vector_type(2))) float v2f;
typedef __attribute__((ext_vector_type(8))) float v8f;

#define NEG_SLOPE 0.2f
#define LN_EPS 1e-5f

static __device__ inline v8f wmma4(v2f a, v2f b, v8f c) {
  // D = A(16x4,f32) * B(4x16,f32) + C(16x16,f32)
  return __builtin_amdgcn_wmma_f32_16x16x4_f32(
      /*neg_a=*/false, a, /*neg_b=*/false, b,
      /*c_mod=*/(short)0, c, /*reuse_a=*/false, /*reuse_b=*/false);
}

// float atomic max via int/uint trick (works with -inf init)
static __device__ inline void atomicMaxF(float* addr, float val) {
  if (val >= 0.f) atomicMax((int*)addr, __float_as_int(val));
  else            atomicMin((unsigned int*)addr, __float_as_uint(val));
}

__global__ void fill_f32(float* __restrict__ p, long long n, float v) {
  long long i = (long long)blockIdx.x * blockDim.x + threadIdx.x;
  if (i < n) p[i] = v;
}

// ---------------- Layer 1 GEMM: [n,4] x [4,64] (x2 weights) ----------------
// One wave per 16-node tile. K=4 exactly -> one WMMA per 16x16 output tile.
__global__ void gemm1_wmma(const float* __restrict__ x,
                           const float* __restrict__ Wl,
                           const float* __restrict__ Wr,
                           float* __restrict__ xl,
                           float* __restrict__ xr, int ntiles) {
  int wave = (int)(((long long)blockIdx.x * blockDim.x + threadIdx.x) >> 5);
  int lane = threadIdx.x & 31;
  if (wave >= ntiles) return;            // wave-uniform: EXEC stays all-ones
  int row0 = wave * 16;
  int m  = lane & 15;
  int hi = lane >> 4;                    // 0: K=0,1  1: K=2,3
  // A tile: 16 nodes x 4 features
  const float* xrow = x + (size_t)(row0 + m) * 4 + hi * 2;
  v2f a; a.x = xrow[0]; a.y = xrow[1];
#pragma unroll
  for (int j = 0; j < 4; ++j) {          // 4 column tiles of 16 (64 out ch)
    v2f bl, br;
    bl.x = Wl[(2 * hi + 0) * 64 + j * 16 + m];
    bl.y = Wl[(2 * hi + 1) * 64 + j * 16 + m];
    br.x = Wr[(2 * hi + 0) * 64 + j * 16 + m];
    br.y = Wr[(2 * hi + 1) * 64 + j * 16 + m];
    v8f cl = {}; cl = wmma4(a, bl, cl);
    v8f cr = {}; cr = wmma4(a, br, cr);
#pragma unroll
    for (int v = 0; v < 8; ++v) {
      int r = v + hi * 8;
      xl[(size_t)(row0 + r) * 64 + j * 16 + m] = cl[v];
      xr[(size_t)(row0 + r) * 64 + j * 16 + m] = cr[v];
    }
  }
}

// ---------------- Layer 2 GEMM: [n,64] x [64,32] (x2 weights) --------------
__global__ void gemm2_wmma(const float* __restrict__ h,
                           const float* __restrict__ Wl,
                           const float* __restrict__ Wr,
                           float* __restrict__ xl,
                           float* __restrict__ xr, int ntiles) {
  int wave = (int)(((long long)blockIdx.x * blockDim.x + threadIdx.x) >> 5);
  int lane = threadIdx.x & 31;
  if (wave >= ntiles) return;
  int row0 = wave * 16;
  int m  = lane & 15;
  int hi = lane >> 4;
  const float* hrow = h + (size_t)(row0 + m) * 64;
  v8f cl[2] = {{0.f},{0.f}};
  v8f cr[2] = {{0.f},{0.f}};
#pragma unroll
  for (int kk = 0; kk < 16; ++kk) {      // K=64 in steps of 4
    int k0 = kk * 4 + hi * 2;
    v2f a; a.x = hrow[k0]; a.y = hrow[k0 + 1];
#pragma unroll
    for (int j = 0; j < 2; ++j) {        // 2 column tiles (32 out ch)
      v2f bl, br;
      bl.x = Wl[(k0 + 0) * 32 + j * 16 + m];
      bl.y = Wl[(k0 + 1) * 32 + j * 16 + m];
      br.x = Wr[(k0 + 0) * 32 + j * 16 + m];
      br.y = Wr[(k0 + 1) * 32 + j * 16 + m];
      cl[j] = wmma4(a, bl, cl[j]);
      cr[j] = wmma4(a, br, cr[j]);
    }
  }
#pragma unroll
  for (int j = 0; j < 2; ++j)
#pragma unroll
    for (int v = 0; v < 8; ++v) {
      int r = v + hi * 8;
      xl[(size_t)(row0 + r) * 32 + j * 16 + m] = cl[j][v];
      xr[(size_t)(row0 + r) * 32 + j * 16 + m] = cr[j][v];
    }
}

// ------------- Layer 1 edge pass A: logits + segment max -------------------
__global__ void edge_logits1(const int* __restrict__ ei, int E, int n,
                             const float* __restrict__ xl,
                             const float* __restrict__ xr,
                             const float* __restrict__ a1,
                             float* __restrict__ logits,
                             float* __restrict__ maxd) {
  int t = blockIdx.x * blockDim.x + threadIdx.x;
  int TE = E + n;
  if (t >= TE) return;
  int s, d;
  if (t < E) { s = ei[t]; d = ei[E + t]; } else { s = t - E; d = s; }
  const float4* ps = (const float4*)(xl + (size_t)s * 64);
  const float4* pd = (const float4*)(xr + (size_t)d * 64);
  const float4* pa = (const float4*)a1;
#pragma unroll
  for (int h = 0; h < 4; ++h) {
    float acc = 0.f;
#pragma unroll
    for (int q = 0; q < 4; ++q) {
      float4 u = ps[h * 4 + q], v = pd[h * 4 + q], w = pa[h * 4 + q];
      float e0 = u.x + v.x; e0 = e0 >= 0.f ? e0 : NEG_SLOPE * e0;
      float e1 = u.y + v.y; e1 = e1 >= 0.f ? e1 : NEG_SLOPE * e1;
      float e2 = u.z + v.z; e2 = e2 >= 0.f ? e2 : NEG_SLOPE * e2;
      float e3 = u.w + v.w; e3 = e3 >= 0.f ? e3 : NEG_SLOPE * e3;
      acc += e0 * w.x + e1 * w.y + e2 * w.z + e3 * w.w;
    }
    logits[(size_t)t * 4 + h] = acc;
    atomicMaxF(&maxd[(size_t)d * 4 + h], acc);
  }
}

// ------- Layer 1 edge pass B: exp, denom-sum, weighted-feature-sum ---------
// One thread per (edge, head).
__global__ void edge_accum1(const int* __restrict__ ei, int E, int n,
                            const float* __restrict__ xl,
                            const float* __restrict__ logits,
                            const float* __restrict__ maxd,
                            float* __restrict__ denom,
                            float* __restrict__ acc) {
  int t = blockIdx.x * blockDim.x + threadIdx.x;
  int T = 4 * (E + n);
  if (t >= T) return;
  int e = t >> 2, h = t & 3;
  int s, d;
  if (e < E) { s = ei[e]; d = ei[E + e]; } else { s = e - E; d = s; }
  float ex = __expf(logits[(size_t)e * 4 + h] - maxd[(size_t)d * 4 + h]);
  atomicAdd(&denom[(size_t)d * 4 + h], ex);
  const float* px = xl + (size_t)s * 64 + h * 16;
  float* pa = acc + (size_t)d * 64 + h * 16;
#pragma unroll
  for (int c = 0; c < 16; ++c) atomicAdd(&pa[c], ex * px[c]);
}

// ---- Layer 1 node finalize: acc/denom + bias -> LayerNorm(64) -> ELU ------
// One wave per node, 2 channels per lane, shuffle reductions (wave32).
__global__ void node_fin1(const float* __restrict__ acc,
                          const float* __restrict__ denom,
                          const float* __restrict__ b1,
                          const float* __restrict__ g,
                          const float* __restrict__ be,
                          float* __restrict__ h1, int n) {
  int node = (int)(((long long)blockIdx.x * blockDim.x + threadIdx.x) >> 5);
  int lane = threadIdx.x & 31;
  if (node >= n) return;
  int c0 = lane, c1 = lane + 32;
  float v0 = acc[(size_t)node * 64 + c0] / denom[(size_t)node * 4 + (c0 >> 4)] + b1[c0];
  float v1 = acc[(size_t)node * 64 + c1] / denom[(size_t)node * 4 + (c1 >> 4)] + b1[c1];
  float s = v0 + v1;
#pragma unroll
  for (int o = 16; o; o >>= 1) s += __shfl_xor(s, o, 32);
  float mu = s * (1.f / 64.f);
  float d0 = v0 - mu, d1 = v1 - mu;
  float q = d0 * d0 + d1 * d1;
#pragma unroll
  for (int o = 16; o; o >>= 1) q += __shfl_xor(q, o, 32);
  float rs = rsqrtf(q * (1.f / 64.f) + LN_EPS);
  float y0 = d0 * rs * g[c0] + be[c0];
  float y1 = d1 * rs * g[c1] + be[c1];
  y0 = y0 > 0.f ? y0 : expm1f(y0);       // ELU
  y1 = y1 > 0.f ? y1 : expm1f(y1);
  h1[(size_t)node * 64 + c0] = y0;
  h1[(size_t)node * 64 + c1] = y1;
}

// ------------- Layer 2 edge pass A: logits + segment max (heads=1) ---------
__global__ void edge_logits2(const int* __restrict__ ei, int E, int n,
                             const float* __restrict__ xl,
                             const float* __restrict__ xr,
                             const float* __restrict__ a2,
                             float* __restrict__ logits,
                             float* __restrict__ maxd) {
  int t = blockIdx.x * blockDim.x + threadIdx.x;
  int TE = E + n;
  if (t >= TE) return;
  int s, d;
  if (t < E) { s = ei[t]; d = ei[E + t]; } else { s = t - E; d = s; }
  const float4* ps = (const float4*)(xl + (size_t)s * 32);
  const float4* pd = (const float4*)(xr + (size_t)d * 32);
  const float4* pa = (const float4*)a2;
  float acc = 0.f;
#pragma unroll
  for (int q = 0; q < 8; ++q) {
    float4 u = ps[q], v = pd[q], w = pa[q];
    float e0 = u.x + v.x; e0 = e0 >= 0.f ? e0 : NEG_SLOPE * e0;
    float e1 = u.y + v.y; e1 = e1 >= 0.f ? e1 : NEG_SLOPE * e1;
    float e2 = u.z + v.z; e2 = e2 >= 0.f ? e2 : NEG_SLOPE * e2;
    float e3 = u.w + v.w; e3 = e3 >= 0.f ? e3 : NEG_SLOPE * e3;
    acc += e0 * w.x + e1 * w.y + e2 * w.z + e3 * w.w;
  }
  logits[t] = acc;
  atomicMaxF(&maxd[d], acc);
}

// ------- Layer 2 edge pass B -----------------------------------------------
__global__ void edge_accum2(const int* __restrict__ ei, int E, int n,
                            const float* __restrict__ xl,
                            const float* __restrict__ logits,
                            const float* __restrict__ maxd,
                            float* __restrict__ denom,
                            float* __restrict__ acc) {
  int t = blockIdx.x * blockDim.x + threadIdx.x;
  int TE = E + n;
  if (t >= TE) return;
  int s, d;
  if (t < E) { s = ei[t]; d = ei[E + t]; } else { s = t - E; d = s; }
  float ex = __expf(logits[t] - maxd[d]);
  atomicAdd(&denom[d], ex);
  const float* px = xl + (size_t)s * 32;
  float* pa = acc + (size_t)d * 32;
#pragma unroll
  for (int c = 0; c < 32; ++c) atomicAdd(&pa[c], ex * px[c]);
}

// ---- Layer 2 node finalize: acc/denom + bias -> LayerNorm(32) -> out ------
__global__ void node_fin2(const float* __restrict__ acc,
                          const float* __restrict__ denom,
                          const float* __restrict__ b2,
                          const float* __restrict__ g,
                          const float* __restrict__ be,
                          float* __restrict__ out, int n) {
  int node = (int)(((long long)blockIdx.x * blockDim.x + threadIdx.x) >> 5);
  int lane = threadIdx.x & 31;
  if (node >= n) return;
  float v = acc[(size_t)node * 32 + lane] / denom[node] + b2[lane];
  float s = v;
#pragma unroll
  for (int o = 16; o; o >>= 1) s += __shfl_xor(s, o, 32);
  float mu = s * (1.f / 32.f);
  float dd = v - mu;
  float q = dd * dd;
#pragma unroll
  for (int o = 16; o; o >>= 1) q += __shfl_xor(q, o, 32);
  float rs = rsqrtf(q * (1.f / 32.f) + LN_EPS);
  out[(size_t)node * 32 + lane] = dd * rs * g[lane] + be[lane];
}

// ---------------------------------------------------------------------------
static inline int gridFor(long long threads, int block) {
  return (int)((threads + block - 1) / block);
}

extern "C" void kernel_launch(void* const* d_in, const int* in_sizes, int n_in,
                              void* d_out, int out_size, void* d_ws, size_t ws_size,
                              hipStream_t stream) {
  const float* x   = (const float*)d_in[0];
  const int*   ei  = (const int*)  d_in[1];
  const float* W1l = (const float*)d_in[2];
  const float* W1r = (const float*)d_in[3];
  const float* a1  = (const float*)d_in[4];
  const float* b1  = (const float*)d_in[5];
  const float* g1  = (const float*)d_in[6];
  const float* be1 = (const float*)d_in[7];
  const float* W2l = (const float*)d_in[8];
  const float* W2r = (const float*)d_in[9];
  const float* a2  = (const float*)d_in[10];
  const float* b2  = (const float*)d_in[11];
  const float* g2  = (const float*)d_in[12];
  const float* be2 = (const float*)d_in[13];
  float* out = (float*)d_out;

  const int n  = in_sizes[0] / 4;     // 100000 (divisible by 16)
  const int E  = in_sizes[1] / 2;     // 1600000
  const int TE = E + n;               // edges incl. self-loops
  const int ntiles = n / 16;

  // scratch layout (floats); layer-2 buffers alias dead layer-1 buffers
  float* ws = (float*)d_ws;
  const size_t XL1  = 0;
  const size_t XR1  = XL1 + (size_t)n * 64;
  const size_t ACC1 = XR1 + (size_t)n * 64;
  const size_t DEN1 = ACC1 + (size_t)n * 64;   // contiguous after ACC1
  const size_t MAX1 = DEN1 + (size_t)n * 4;
  const size_t LOG1 = MAX1 + (size_t)n * 4;
  const size_t H1   = LOG1 + (size_t)TE * 4;
  // aliases: XL2=XL1, XR2=XR1, ACC2=ACC1, DEN2=DEN1, MAX2=MAX1, LOG2=LOG1

  const int B = 256;
  const float NEG_INF = -INFINITY;

  // ---- layer 1 ----
  fill_f32<<<gridFor((long long)n * 68, B), B, 0, stream>>>(ws + ACC1, (long long)n * 68, 0.f);
  fill_f32<<<gridFor((long long)n * 4, B), B, 0, stream>>>(ws + MAX1, (long long)n * 4, NEG_INF);

  gemm1_wmma<<<gridFor((long long)ntiles * 32, B), B, 0, stream>>>(
      x, W1l, W1r, ws + XL1, ws + XR1, ntiles);

  edge_logits1<<<gridFor(TE, B), B, 0, stream>>>(
      ei, E, n, ws + XL1, ws + XR1, a1, ws + LOG1, ws + MAX1);

  edge_accum1<<<gridFor((long long)TE * 4, B), B, 0, stream>>>(
      ei, E, n, ws + XL1, ws + LOG1, ws + MAX1, ws + DEN1, ws + ACC1);

  node_fin1<<<gridFor((long long)n * 32, B), B, 0, stream>>>(
      ws + ACC1, ws + DEN1, b1, g1, be1, ws + H1, n);

  // ---- layer 2 (reuse layer-1 regions) ----
  fill_f32<<<gridFor((long long)n * 32, B), B, 0, stream>>>(ws + ACC1, (long long)n * 32, 0.f);
  fill_f32<<<gridFor(n, B), B, 0, stream>>>(ws + DEN1, n, 0.f);
  fill_f32<<<gridFor(n, B), B, 0, stream>>>(ws + MAX1, n, NEG_INF);

  gemm2_wmma<<<gridFor((long long)ntiles * 32, B), B, 0, stream>>>(
      ws + H1, W2l, W2r, ws + XL1, ws + XR1, ntiles);

  edge_logits2<<<gridFor(TE, B), B, 0, stream>>>(
      ei, E, n, ws + XL1, ws + XR1, a2, ws + LOG1, ws + MAX1);

  edge_accum2<<<gridFor(TE, B), B, 0, stream>>>(
      ei, E, n, ws + XL1, ws + LOG1, ws + MAX1, ws + DEN1, ws + ACC1);

  node_fin2<<<gridFor((long long)n * 32, B), B, 0, stream>>>(
      ws + ACC1, ws + DEN1, b2, g2, be2, out, n);
}